// EfficientMemorySiLU_52183852646965
// MI455X (gfx1250) — compile-verified
//
#include <hip/hip_runtime.h>

// SiLU forward: y = x * sigmoid(x) = x / (1 + exp(-x))
// 4 x 4096 x 4096 fp32 -> purely HBM-bandwidth bound on MI455X (23.3 TB/s,
// 512 MB traffic => ~22 us floor). Optimal shape: clause of 8 NT b128 loads,
// v_exp_f32 / v_rcp_f32 sigmoid, clause of 8 NT b128 stores. Bounds check is
// uniform (SALU) per block; only the last ragged block diverges.

typedef float v4f __attribute__((ext_vector_type(4)));

constexpr unsigned BLOCK  = 256;  // 8 wave32s per block
constexpr unsigned UNROLL = 8;    // 8 x float4 = 32 elements per thread

__device__ __forceinline__ float silu1(float x) {
    float e = __expf(-x);                       // v_exp_f32 (TRANS32)
    return x * __builtin_amdgcn_rcpf(1.0f + e); // v_rcp_f32 (TRANS32) + v_mul
}

__device__ __forceinline__ v4f silu4(v4f t) {
    v4f r;
    r.x = silu1(t.x);
    r.y = silu1(t.y);
    r.z = silu1(t.z);
    r.w = silu1(t.w);
    return r;
}

__global__ __launch_bounds__(BLOCK) void silu_vec4_kernel(
    const v4f* __restrict__ x, v4f* __restrict__ y, unsigned n4) {
    unsigned base     = blockIdx.x * (BLOCK * UNROLL) + threadIdx.x;
    unsigned tile_end = (blockIdx.x + 1u) * (BLOCK * UNROLL);

    if (tile_end <= n4) {
        // Fast path (all blocks except possibly the last): no per-lane guards.
        // 8 independent NT b128 loads -> single clause, deep MLP.
        v4f v[UNROLL];
#pragma unroll
        for (unsigned k = 0; k < UNROLL; ++k) {
            v[k] = __builtin_nontemporal_load(x + base + k * BLOCK);
        }
#pragma unroll
        for (unsigned k = 0; k < UNROLL; ++k) {
            v[k] = silu4(v[k]);
        }
#pragma unroll
        for (unsigned k = 0; k < UNROLL; ++k) {
            __builtin_nontemporal_store(v[k], y + base + k * BLOCK);
        }
    } else {
        // Ragged last block: per-chunk guard, load-compute-store to keep
        // register pressure and control flow trivial.
#pragma unroll
        for (unsigned k = 0; k < UNROLL; ++k) {
            unsigned i = base + k * BLOCK;
            if (i < n4) {
                v4f t = __builtin_nontemporal_load(x + i);
                __builtin_nontemporal_store(silu4(t), y + i);
            }
        }
    }
}

// Scalar tail for element counts not divisible by 4 (not hit for this shape).
__global__ __launch_bounds__(64) void silu_tail_kernel(
    const float* __restrict__ x, float* __restrict__ y,
    unsigned start, unsigned n) {
    unsigned i = start + blockIdx.x * blockDim.x + threadIdx.x;
    if (i < n) {
        y[i] = silu1(x[i]);
    }
}

extern "C" void kernel_launch(void* const* d_in, const int* in_sizes, int n_in,
                              void* d_out, int out_size, void* d_ws, size_t ws_size,
                              hipStream_t stream) {
    (void)n_in; (void)out_size; (void)d_ws; (void)ws_size;

    const float* x = (const float*)d_in[0];
    float*       y = (float*)d_out;
    unsigned n   = (unsigned)in_sizes[0];   // 67,108,864 for this shape
    unsigned n4  = n / 4u;                  // 16,777,216 float4s
    unsigned rem = n - n4 * 4u;

    if (n4 > 0) {
        unsigned per_block = BLOCK * UNROLL;                  // 2048 float4s/block
        unsigned blocks = (n4 + per_block - 1u) / per_block;  // 8192 blocks
        silu_vec4_kernel<<<blocks, BLOCK, 0, stream>>>(
            (const v4f*)x, (v4f*)y, n4);
    }
    if (rem > 0) {
        silu_tail_kernel<<<1, 64, 0, stream>>>(x, y, n4 * 4u, n);
    }
}